// GAT_13623636262977
// MI455X (gfx1250) — compile-verified
//
#include <hip/hip_runtime.h>
#include <hip/hip_bf16.h>

// ---------------- problem constants ----------------
#define NN 50000          // nodes
#define NE 800000         // real edges
#define ET 850000         // edges + self loops
#define NEG_SLOPE 0.2f

// ---------------- WMMA types ----------------
typedef __attribute__((ext_vector_type(2))) float v2f;
typedef __attribute__((ext_vector_type(8))) float v8f;

// ---------------- fp32 WMMA GEMM: H = X @ W ----------------
// block = 256 threads (8 waves). Tile: TM x TN = 128 x 64 per block,
// each wave owns 16 rows x 64 cols (4 col-tiles of 16x16).
// K staged through LDS in KC=32 chunks, WMMA f32 16x16x4 steps.
// A and B fragments are stored so each lane's (x,y) pair is one contiguous
// 8B-aligned LDS pair -> single ds_load_b64 into an even VGPR pair, no movs.
#define TM 128
#define TN 64
#define KC 32
#define APAD 2            // row stride 34 dwords: 8B-aligned pairs, distinct banks

__global__ __launch_bounds__(256) void gemm_wmma_f32(
    const float* __restrict__ X, const float* __restrict__ W,
    float* __restrict__ H, int M, int K, int N)
{
    __shared__ float As[TM][KC + APAD];   // row-major rows of X
    __shared__ float Bs[TN][KC + APAD];   // TRANSPOSED: Bs[col][k]

    const int tid  = threadIdx.x;
    const int wave = tid >> 5;
    const int lane = tid & 31;
    const int rowBase = blockIdx.x * TM;
    const int colBase = blockIdx.y * TN;

    v8f acc[TN / 16];
#pragma unroll
    for (int t = 0; t < TN / 16; ++t)
        acc[t] = (v8f){0.f, 0.f, 0.f, 0.f, 0.f, 0.f, 0.f, 0.f};

    for (int k0 = 0; k0 < K; k0 += KC) {
        // stage A tile: TM x KC (float4 global loads, scalar LDS stores)
        for (int i = tid; i < TM * (KC / 4); i += 256) {
            int r = i >> 3;              // KC/4 = 8 quads per row
            int q = (i & 7) << 2;
            int row = rowBase + r;
            float4 v = make_float4(0.f, 0.f, 0.f, 0.f);
            if (row < M) v = *(const float4*)(X + (size_t)row * K + k0 + q);
            As[r][q + 0] = v.x; As[r][q + 1] = v.y;
            As[r][q + 2] = v.z; As[r][q + 3] = v.w;
        }
        // stage B tile transposed: read W[k][col] row-major, store Bs[col][k]
        for (int i = tid; i < KC * (TN / 4); i += 256) {
            int r = i >> 4;              // k within chunk
            int q = (i & 15) << 2;       // starting col quad
            float4 v = *(const float4*)(W + (size_t)(k0 + r) * N + colBase + q);
            Bs[q + 0][r] = v.x; Bs[q + 1][r] = v.y;
            Bs[q + 2][r] = v.z; Bs[q + 3][r] = v.w;
        }
        __syncthreads();

        // A 16x4 fragment: lanes 0-15 row M=lane (K=k,k+1), lanes 16-31 same rows (K=k+2,k+3)
        // B 4x16 fragment: symmetric; with transposed Bs both are contiguous pairs.
        const int arow = (wave << 4) + (lane & 15);
        const int kh   = (lane >> 4) << 1;     // 0 or 2
        const int bc   = lane & 15;
#pragma unroll
        for (int kk = 0; kk < KC; kk += 4) {
            v2f a = *(const v2f*)&As[arow][kk + kh];
#pragma unroll
            for (int t = 0; t < TN / 16; ++t) {
                v2f b = *(const v2f*)&Bs[(t << 4) + bc][kk + kh];
                acc[t] = __builtin_amdgcn_wmma_f32_16x16x4_f32(
                    false, a, false, b, (short)0, acc[t], false, false);
            }
        }
        __syncthreads();
    }

    // C/D layout: VGPR i -> M=i (lanes 0-15), M=i+8 (lanes 16-31); N = lane&15
    const int outRow0 = rowBase + (wave << 4) + ((lane >> 4) << 3);
    const int col     = colBase + (lane & 15);
#pragma unroll
    for (int t = 0; t < TN / 16; ++t) {
        int c = col + (t << 4);
#pragma unroll
        for (int i = 0; i < 8; ++i) {
            int row = outRow0 + i;
            if (row < M) H[(size_t)row * N + c] = acc[t][i];
        }
    }
}

// ---------------- attention helpers ----------------
__device__ __forceinline__ void edge_nodes(const int* __restrict__ ei, int e, int& s, int& d) {
    if (e < NE) { s = ei[e]; d = ei[NE + e]; }
    else        { s = e - NE; d = e - NE; }           // synthesized self-loop
}

// monotone float -> uint mapping so unsigned atomicMax orders like float max
__device__ __forceinline__ unsigned fmap(float f) {
    unsigned u = __float_as_uint(f);
    return (u & 0x80000000u) ? ~u : (u | 0x80000000u);
}
__device__ __forceinline__ float funmap(unsigned m) {
    unsigned u = (m & 0x80000000u) ? (m ^ 0x80000000u) : ~m;
    return __uint_as_float(u);
}

// per-(node, head): alpha_src / alpha_dst dot products over C channels
__global__ void node_alpha_k(const float* __restrict__ H,
                             const float* __restrict__ a_s, const float* __restrict__ a_d,
                             float* __restrict__ als, float* __restrict__ ald,
                             int heads, int C)
{
    int i = blockIdx.x * blockDim.x + threadIdx.x;
    if (i >= NN * heads) return;
    int n = i / heads, h = i - n * heads;
    const float* hp = H + (size_t)n * heads * C + (size_t)h * C;
    float s = 0.f, d = 0.f;
    for (int c = 0; c < C; ++c) {
        float v = hp[c];
        s = fmaf(v, a_s[h * C + c], s);
        d = fmaf(v, a_d[h * C + c], d);
    }
    als[i] = s; ald[i] = d;
}

// per-(edge, head): segment max of leaky_relu(al_s[src]+al_d[dst]) over dst
__global__ void edge_max_k(const int* __restrict__ ei,
                           const float* __restrict__ als, const float* __restrict__ ald,
                           unsigned* __restrict__ mmax, int heads)
{
    int i = blockIdx.x * blockDim.x + threadIdx.x;
    if (i >= ET * heads) return;
    int e = i / heads, h = i - e * heads;
    int s, d; edge_nodes(ei, e, s, d);
    float ev = als[s * heads + h] + ald[d * heads + h];
    ev = ev > 0.f ? ev : NEG_SLOPE * ev;
    atomicMax(&mmax[d * heads + h], fmap(ev));
}

// per-(edge, head): exp(e - max) and denominator accumulation
__global__ void edge_exp_k(const int* __restrict__ ei,
                           const float* __restrict__ als, const float* __restrict__ ald,
                           const unsigned* __restrict__ mmax,
                           float* __restrict__ EX, float* __restrict__ DEN, int heads)
{
    int i = blockIdx.x * blockDim.x + threadIdx.x;
    if (i >= ET * heads) return;
    int e = i / heads, h = i - e * heads;
    int s, d; edge_nodes(ei, e, s, d);
    float ev = als[s * heads + h] + ald[d * heads + h];
    ev = ev > 0.f ? ev : NEG_SLOPE * ev;
    float m  = funmap(mmax[d * heads + h]);
    float ex = __expf(ev - m);
    EX[i] = ex;
    atomicAdd(&DEN[d * heads + h], ex);
}

// one block per edge, heads*C threads over channels: weighted scatter-add
__global__ void edge_aggr_k(const float* __restrict__ H,
                            const float* __restrict__ EX, const float* __restrict__ DEN,
                            const int* __restrict__ ei, float* __restrict__ agg,
                            int heads, int C)
{
    int e = blockIdx.x;
    int c = threadIdx.x;            // 0 .. heads*C-1
    int h = c / C;
    int s, d; edge_nodes(ei, e, s, d);
    float alpha = EX[(size_t)e * heads + h] / DEN[d * heads + h];
    atomicAdd(&agg[(size_t)d * heads * C + c],
              alpha * H[(size_t)s * heads * C + c]);
}

// bias (+ optional ReLU, optional residual)
__global__ void finalize_k(const float* __restrict__ agg, const float* __restrict__ bias,
                           const float* __restrict__ resid, float* __restrict__ out,
                           long total, int HC, int do_relu)
{
    long i = blockIdx.x * (long)blockDim.x + threadIdx.x;
    if (i >= total) return;
    float v = agg[i] + bias[(int)(i % HC)];
    if (do_relu) v = fmaxf(v, 0.f);
    if (resid)   v += resid[i];
    out[i] = v;
}

// ---------------- host-side per-layer driver ----------------
static void run_layer(const float* Xin, const float* Wm,
                      const float* a_s, const float* a_d, const int* ei,
                      float* H, float* AGG, float* EX, float* ALS, float* ALD,
                      unsigned* MMAX, float* DEN,
                      int K, int heads, int C, hipStream_t stream)
{
    const int N = heads * C;
    dim3 gg((NN + TM - 1) / TM, N / TN);
    gemm_wmma_f32<<<gg, 256, 0, stream>>>(Xin, Wm, H, NN, K, N);

    hipMemsetAsync(MMAX, 0, sizeof(unsigned) * (size_t)NN * heads, stream); // mapped -inf
    hipMemsetAsync(DEN,  0, sizeof(float)    * (size_t)NN * heads, stream);
    hipMemsetAsync(AGG,  0, sizeof(float)    * (size_t)NN * N,     stream);

    node_alpha_k<<<(NN * heads + 255) / 256, 256, 0, stream>>>(H, a_s, a_d, ALS, ALD, heads, C);
    edge_max_k  <<<(ET * heads + 255) / 256, 256, 0, stream>>>(ei, ALS, ALD, MMAX, heads);
    edge_exp_k  <<<(ET * heads + 255) / 256, 256, 0, stream>>>(ei, ALS, ALD, MMAX, EX, DEN, heads);
    edge_aggr_k <<<ET, N, 0, stream>>>(H, EX, DEN, ei, AGG, heads, C);
}

extern "C" void kernel_launch(void* const* d_in, const int* in_sizes, int n_in,
                              void* d_out, int out_size, void* d_ws, size_t ws_size,
                              hipStream_t stream)
{
    const float* x   = (const float*)d_in[0];
    const int*   ei  = (const int*)  d_in[1];   // (2, NE) int32 row-major
    const float* W1  = (const float*)d_in[2];
    const float* as1 = (const float*)d_in[3];
    const float* ad1 = (const float*)d_in[4];
    const float* b1  = (const float*)d_in[5];
    const float* W2  = (const float*)d_in[6];
    const float* as2 = (const float*)d_in[7];
    const float* ad2 = (const float*)d_in[8];
    const float* b2  = (const float*)d_in[9];
    const float* W3  = (const float*)d_in[10];
    const float* as3 = (const float*)d_in[11];
    const float* ad3 = (const float*)d_in[12];
    const float* b3  = (const float*)d_in[13];
    float* out = (float*)d_out;

    // workspace carve-up (floats)
    const size_t NB = (size_t)NN * 256;           // node-feature buffer
    float*    H    = (float*)d_ws;                // [NN,256]
    float*    AGG  = H   + NB;                    // [NN,256]
    float*    X1   = AGG + NB;                    // [NN,256]
    float*    EX   = X1  + NB;                    // [ET,4]
    float*    ALS  = EX  + (size_t)ET * 4;        // [NN,4]
    float*    ALD  = ALS + (size_t)NN * 4;        // [NN,4]
    unsigned* MMAX = (unsigned*)(ALD + (size_t)NN * 4);   // [NN,4]
    float*    DEN  = (float*)(MMAX + (size_t)NN * 4);     // [NN,4]

    const long NB256 = (long)NN * 256;
    const long NB64  = (long)NN * 64;

    // ---- layer 1: x[128] -> h1[4x64], concat, ReLU ----
    run_layer(x, W1, as1, ad1, ei, H, AGG, EX, ALS, ALD, MMAX, DEN, 128, 4, 64, stream);
    finalize_k<<<(NB256 + 255) / 256, 256, 0, stream>>>(AGG, b1, nullptr, X1, NB256, 256, 1);

    // ---- layer 2: x1[256] -> h2[4x64], concat, ReLU, + x1 residual ----
    run_layer(X1, W2, as2, ad2, ei, H, AGG, EX, ALS, ALD, MMAX, DEN, 256, 4, 64, stream);
    finalize_k<<<(NB256 + 255) / 256, 256, 0, stream>>>(AGG, b2, X1, AGG, NB256, 256, 1);  // x2 in AGG

    // ---- layer 3: x2[256] -> h3[1x64], mean over 1 head == identity ----
    run_layer(AGG, W3, as3, ad3, ei, H, X1 /*agg target*/, EX, ALS, ALD, MMAX, DEN, 256, 1, 64, stream);
    finalize_k<<<(NB64 + 255) / 256, 256, 0, stream>>>(X1, b3, nullptr, out, NB64, 64, 0);
}